// SELayer_49503793054539
// MI455X (gfx1250) — compile-verified
//
#include <hip/hip_runtime.h>

// Native clang ext-vectors (work with __builtin_nontemporal_* and WMMA builtins)
typedef float f32x4 __attribute__((ext_vector_type(4)));
typedef float v2f   __attribute__((ext_vector_type(2)));
typedef float v8f   __attribute__((ext_vector_type(8)));

#define HW_   16384              // 128*128
#define C_    256
#define CHW_  (C_ * HW_)         // elements in one batch image: 4,194,304
#define HID_  16

// ---------------------------------------------------------------------------
// Kernel 1: global average pool of batch element 0 only.
// One block per channel; 256 threads * 16 float4 loads = 16384 elements.
// ---------------------------------------------------------------------------
__global__ void se_pool(const float* __restrict__ x, float* __restrict__ ws_y) {
    __shared__ float red[256];
    const int c = blockIdx.x;
    const int t = threadIdx.x;
    const f32x4* xc = reinterpret_cast<const f32x4*>(x + (size_t)c * HW_);
    float s = 0.f;
#pragma unroll
    for (int i = 0; i < 16; ++i) {
        f32x4 v = xc[t + 256 * i];     // coalesced B128 loads
        s += v.x + v.y + v.z + v.w;
    }
    red[t] = s;
    __syncthreads();
#pragma unroll
    for (int off = 128; off > 0; off >>= 1) {
        if (t < off) red[t] += red[t + off];
        __syncthreads();
    }
    if (t == 0) ws_y[c] = red[0] * (1.0f / 16384.0f);
}

// ---------------------------------------------------------------------------
// Kernel 2: the 256->16->256 MLP on a single wave using V_WMMA_F32_16X16X4_F32.
// fp32 WMMA keeps full precision vs the fp32 reference.
//
// ISA 32-bit layouts (CDNA5 §7.12.2):
//   A (16x4, MxK): lanes 0-15 hold M=lane, VGPR0=K0,VGPR1=K1;
//                  lanes 16-31 hold M=lane-16, VGPR0=K2,VGPR1=K3.
//   B (4x16, KxN): lanes 0-15 hold N=lane with K0/K1 in the 2 VGPRs;
//                  lanes 16-31 hold N=lane-16 with K2/K3.
//   D (16x16):     row M=0 lives in acc[0] of lanes 0-15 (N=lane).
// Only row 0 of the A matrices is nonzero (batch 0).
//
// All loads are UNCONDITIONAL; the "row 0 only" mask is applied with a
// select afterwards so it lowers to v_cndmask instead of exec-masked loads.
// ---------------------------------------------------------------------------
__global__ void se_mlp(const float* __restrict__ w1, const float* __restrict__ b1,
                       const float* __restrict__ w2, const float* __restrict__ b2,
                       const float* __restrict__ ws_y, float* __restrict__ ws_s) {
    __shared__ float ly[C_];        // staged pooled means
    __shared__ float hbuf[HID_];    // hidden activations
    const int lane = threadIdx.x;   // 0..31, one wave, EXEC all 1s
    const int half = lane >> 4;     // selects K-pair {0,1} vs {2,3}
    const int idx  = lane & 15;     // M for A, N for B/D
    const bool row0 = (idx == 0);

    // Stage y[256] into LDS: 8 floats per lane, coalesced.
#pragma unroll
    for (int i = 0; i < C_ / 32; ++i) ly[lane + 32 * i] = ws_y[lane + 32 * i];
    __syncthreads();

    // ---- GEMM1: D = Y(16x256) * W1^T(256x16), K tiled by 4 (64 WMMA ops)
    v8f acc = {};
    for (int k0 = 0; k0 < C_; k0 += 4) {
        const int kk = k0 + 2 * half;
        const float y0 = ly[kk + 0];               // unconditional ds_load
        const float y1 = ly[kk + 1];
        v2f a, b;
        a.x = row0 ? y0 : 0.f;                     // v_cndmask, no branch
        a.y = row0 ? y1 : 0.f;
        b.x = w1[idx * C_ + kk + 0];               // B[k][n] = w1[n][k] (b64)
        b.y = w1[idx * C_ + kk + 1];
        acc = __builtin_amdgcn_wmma_f32_16x16x4_f32(
            false, a, false, b, (short)0, acc, false, false);
    }
    if (lane < 16) {                               // D row 0 -> hidden, bias+ReLU
        float h = acc[0] + b1[lane];
        hbuf[lane] = h > 0.f ? h : 0.f;
    }
    __syncthreads();

    // ---- GEMM2: S = H(16x16) * W2^T(16x256), 16 N-tiles x 4 K-steps
    for (int t = 0; t < 16; ++t) {
        v8f acc2 = {};
#pragma unroll
        for (int k0 = 0; k0 < HID_; k0 += 4) {
            const int kk = k0 + 2 * half;
            const float h0 = hbuf[kk + 0];         // unconditional ds_load
            const float h1 = hbuf[kk + 1];
            v2f a, b;
            a.x = row0 ? h0 : 0.f;
            a.y = row0 ? h1 : 0.f;
            const int n = t * 16 + idx;            // global output channel
            b.x = w2[n * HID_ + kk + 0];           // B[k][n] = w2[n][k] (b64)
            b.y = w2[n * HID_ + kk + 1];
            acc2 = __builtin_amdgcn_wmma_f32_16x16x4_f32(
                false, a, false, b, (short)0, acc2, false, false);
        }
        if (lane < 16) {
            const int n = t * 16 + lane;
            const float z = acc2[0] + b2[n];
            ws_s[n] = 1.0f / (1.0f + __expf(-z));  // sigmoid gate
        }
    }
}

// ---------------------------------------------------------------------------
// Kernel 3: the bandwidth pass. out = x, with batch 0 scaled per-channel.
// One float4 per thread; non-temporal B128 so the 512 MiB stream doesn't
// thrash L2. 16384 % 4 == 0 so a float4 never straddles a channel boundary,
// and CHW_ is a multiple of the per-block element count (1024), so the
// batch-0 branch is block-uniform.
// ---------------------------------------------------------------------------
__global__ void se_scale(const float* __restrict__ x, const float* __restrict__ ws_s,
                         float* __restrict__ out) {
    const unsigned i4 = blockIdx.x * 256u + threadIdx.x; // float4 index
    const unsigned e  = i4 << 2;                         // element index
    f32x4 v = __builtin_nontemporal_load(reinterpret_cast<const f32x4*>(x) + i4);
    float scale = 1.0f;
    if (e < (unsigned)CHW_) scale = ws_s[(e >> 14) & 255u]; // batch 0: channel gate
    v *= scale;
    __builtin_nontemporal_store(v, reinterpret_cast<f32x4*>(out) + i4);
}

// ---------------------------------------------------------------------------
extern "C" void kernel_launch(void* const* d_in, const int* in_sizes, int n_in,
                              void* d_out, int out_size, void* d_ws, size_t ws_size,
                              hipStream_t stream) {
    (void)in_sizes; (void)n_in; (void)ws_size;
    const float* x  = (const float*)d_in[0];
    const float* w1 = (const float*)d_in[1];
    const float* b1 = (const float*)d_in[2];
    const float* w2 = (const float*)d_in[3];
    const float* b2 = (const float*)d_in[4];
    float* out  = (float*)d_out;
    float* ws_y = (float*)d_ws;        // [256] pooled means
    float* ws_s = ws_y + C_;           // [256] sigmoid gates

    se_pool<<<C_, 256, 0, stream>>>(x, ws_y);
    se_mlp<<<1, 32, 0, stream>>>(w1, b1, w2, b2, ws_y, ws_s);

    const unsigned n4 = (unsigned)out_size / 4u;   // 16,777,216 float4s
    se_scale<<<n4 / 256u, 256, 0, stream>>>(x, ws_s, out);
}